// MeanPooling_2877628088531
// MI455X (gfx1250) — compile-verified
//
#include <hip/hip_runtime.h>
#include <hip/hip_bf16.h>

// ---------------------------------------------------------------------------
// scatter_mean(x[N,512], index[N]) -> out[2048,512]
// Bandwidth-bound (410 MB stream @ 23.3 TB/s => ~18us floor). WMMA is not
// applicable (one-hot GEMM would be ~1e4x the flops); the CDNA5 paths that
// matter here are ds_add_f32 (LDS-privatized accumulation),
// global_atomic_add_f32 (tile flush), non-temporal global_load_b128
// (stream x without evicting the 4MB accumulator set from L2), and
// global_prefetch_b8.
//
// Layout: 32 column tiles x 8 row chunks = 256 workgroups. Each workgroup
// holds a 16-column slice of ALL 2048 segment accumulators in LDS
// (2048*17*4 = 139,264 B < 160 KB CU-mode budget; stride 17 spreads random
// segment IDs over all 64 banks since gcd(17,64)=1). 102M accumulations run
// at LDS rate; only 256*2048*16 = 8.4M f32 atomics touch L2.
// ---------------------------------------------------------------------------

typedef float v4f __attribute__((ext_vector_type(4)));

constexpr int D       = 512;   // feature dim
constexpr int S       = 2048;  // segments
constexpr int COLS    = 16;    // columns per tile
constexpr int PAD     = 17;    // LDS row stride (gcd(17,64)=1 -> bank-spread)
constexpr int CTILES  = D / COLS;        // 32 column tiles
constexpr int RCHUNKS = 8;               // row chunks -> 256 workgroups total
constexpr int BLK     = 256;             // 8 waves of 32

// Fixed launch shapes -> compile-time grid strides (no runtime div codegen).
constexpr int INIT_GRID  = 256;
constexpr int INIT_STRIDE = INIT_GRID * BLK;    // 65536
constexpr int CNT_GRID   = 256;
constexpr int CNT_STRIDE = CNT_GRID * BLK;      // 65536
constexpr int RSTRIDE    = RCHUNKS * (BLK / 4); // 512 rows per grid step

// ---- zero the output accumulators and the counts scratch ------------------
__global__ __launch_bounds__(BLK) void mp_init_kernel(
    float* __restrict__ out, unsigned* __restrict__ counts) {
    const int i = blockIdx.x * BLK + threadIdx.x;
    #pragma unroll
    for (int j = 0; j < (S * D) / INIT_STRIDE; ++j)
        out[i + j * INIT_STRIDE] = 0.0f;     // S*D = 1M = 16 * 65536 exactly
    if (i < S) counts[i] = 0u;
}

// ---- per-segment row counts: LDS histogram (ds_add_u32) + global flush ----
__global__ __launch_bounds__(BLK) void mp_count_kernel(
    const int* __restrict__ index, unsigned* __restrict__ counts, int n) {
    __shared__ unsigned h[S];  // 8 KB
    for (int j = threadIdx.x; j < S; j += BLK) h[j] = 0u;
    __syncthreads();
    const int i = blockIdx.x * BLK + threadIdx.x;
    for (int j = i; j < n; j += CNT_STRIDE) {
        atomicAdd(&h[index[j]], 1u);         // ds_add_u32 (no rtn)
    }
    __syncthreads();
    for (int j = threadIdx.x; j < S; j += BLK) {
        unsigned v = h[j];
        if (v) atomicAdd(&counts[j], v);     // global_atomic_add_u32
    }
}

// ---- main scatter-add: column-tiled LDS accumulation ----------------------
__global__ __launch_bounds__(BLK) void mp_scatter_kernel(
    const float* __restrict__ x, const int* __restrict__ index,
    float* __restrict__ out, int n) {
    __shared__ float acc[S * PAD];  // 139,264 B

    const int ctile = blockIdx.x % CTILES;  // which 16-column slice
    const int chunk = blockIdx.x / CTILES;  // which row chunk
    const int col0  = ctile * COLS;

    for (int j = threadIdx.x; j < S * PAD; j += BLK) acc[j] = 0.0f;
    __syncthreads();

    // 4 lanes per row: each lane owns one float4 (64 B contiguous per quad).
    const int cg   = (threadIdx.x & 3) * 4;  // col offset in tile: 0/4/8/12
    const int rsub = threadIdx.x >> 2;       // 0..63
    const int coff = col0 + cg;

    // One-stage software pipeline: next row's b128 + index loads are issued
    // before the current row's ds_add_f32 chain, keeping a load in flight.
    int r = chunk * (BLK / 4) + rsub;
    if (r < n) {
        v4f v = __builtin_nontemporal_load(
            reinterpret_cast<const v4f*>(&x[(size_t)r * D + coff]));
        int s = index[r];
        for (r += RSTRIDE; r < n; r += RSTRIDE) {
            if (r + 4 * RSTRIDE < n)  // stream hint: global_prefetch_b8
                __builtin_prefetch(&x[(size_t)(r + 4 * RSTRIDE) * D + coff], 0, 0);
            const v4f vn = __builtin_nontemporal_load(
                reinterpret_cast<const v4f*>(&x[(size_t)r * D + coff]));
            const int sn = index[r];

            float* a = &acc[s * PAD + cg];
            atomicAdd(&a[0], v.x);  // ds_add_f32
            atomicAdd(&a[1], v.y);
            atomicAdd(&a[2], v.z);
            atomicAdd(&a[3], v.w);
            v = vn;
            s = sn;
        }
        float* a = &acc[s * PAD + cg];
        atomicAdd(&a[0], v.x);
        atomicAdd(&a[1], v.y);
        atomicAdd(&a[2], v.z);
        atomicAdd(&a[3], v.w);
    }
    __syncthreads();

    // Flush this tile's partial sums: 2048*16 f32 atomics per workgroup.
    #pragma unroll
    for (int k = 0; k < (S * COLS) / BLK; ++k) {
        const int j = k * BLK + threadIdx.x;
        const int s = j >> 4;            // / COLS
        const int c = j & (COLS - 1);
        atomicAdd(&out[s * D + col0 + c], acc[s * PAD + c]);  // global_atomic_add_f32
    }
}

// ---- divide by clamped counts (empty segments stay 0) ---------------------
__global__ __launch_bounds__(BLK) void mp_finalize_kernel(
    float* __restrict__ out, const unsigned* __restrict__ counts) {
    const int i = blockIdx.x * BLK + threadIdx.x;  // grid sized exactly S*D/BLK
    const int s = i >> 9;                          // / 512
    const float c = fmaxf((float)counts[s], 1.0f);
    out[i] = out[i] * __frcp_rn(c);
}

extern "C" void kernel_launch(void* const* d_in, const int* in_sizes, int n_in,
                              void* d_out, int out_size, void* d_ws, size_t ws_size,
                              hipStream_t stream) {
    const float* x      = (const float*)d_in[0];
    const int*   index  = (const int*)d_in[1];  // harness: integer -> const int*
    float*       out    = (float*)d_out;        // [2048, 512]
    unsigned*    counts = (unsigned*)d_ws;      // 8 KB scratch

    const int n = in_sizes[1];  // number of rows (= index element count)

    mp_init_kernel<<<INIT_GRID, BLK, 0, stream>>>(out, counts);
    mp_count_kernel<<<CNT_GRID, BLK, 0, stream>>>(index, counts, n);
    mp_scatter_kernel<<<CTILES * RCHUNKS, BLK, 0, stream>>>(x, index, out, n);
    mp_finalize_kernel<<<(S * D) / BLK, BLK, 0, stream>>>(out, counts);
}